// GNNCritic_7301444403653
// MI455X (gfx1250) — compile-verified
//
#include <hip/hip_runtime.h>
#include <hip/hip_bf16.h>

#define NNODES 50000
#define NEDGES 800000
#define NGRAPHS 64
#define FEAT 128
#define HID 256
#define ACTD 32

// GEMM tiling
#define KC 64            // K-chunk staged in LDS
#define LDSW 80          // padded LDS row stride (floats): +160 fl per 2 rows -> +32 banks

typedef __attribute__((ext_vector_type(2))) float v2f;
typedef __attribute__((ext_vector_type(8))) float v8f;

// ---------------------------------------------------------------------------
// Small utility kernels
// ---------------------------------------------------------------------------
__global__ void k_fill(float* __restrict__ p, float v, int n) {
    int i = blockIdx.x * blockDim.x + threadIdx.x;
    if (i < n) p[i] = v;
}

__global__ void k_deg_edges(const int* __restrict__ dst, float* __restrict__ deg, int e) {
    int i = blockIdx.x * blockDim.x + threadIdx.x;
    if (i < e) atomicAdd(&deg[dst[i]], 1.0f);
}

__global__ void k_rsqrt_inplace(float* __restrict__ deg, int n) {
    int i = blockIdx.x * blockDim.x + threadIdx.x;
    if (i < n) {
        float d = deg[i];
        deg[i] = d > 0.0f ? rsqrtf(d) : 0.0f;
    }
}

__global__ void k_counts(const int* __restrict__ batch, float* __restrict__ counts, int n) {
    int i = blockIdx.x * blockDim.x + threadIdx.x;
    if (i < n) atomicAdd(&counts[batch[i]], 1.0f);
}

// ---------------------------------------------------------------------------
// FP32 WMMA GEMM:  C[M x N] = A[M x K] @ W[K x N]
// Block = 4 waves; each wave -> 16 rows x 64 cols (4 accumulators reuse 1 A frag).
// W chunk (KC x 64) staged cooperatively in LDS, B fragments fed via ds_load.
// Fragment layouts per CDNA5 ISA 7.12.2 (f32 16x16x4):
//   A 16x4 : lane L -> M=L&15, K=(L>>4)*2+v        (v2f)
//   B 4x16 : VGPR v, lanes0-15 K=v, lanes16-31 K=v+2, N=L&15  (v2f)
//   C 16x16: VGPR v -> M=v+(L>>4)*8, N=L&15         (v8f)
// ---------------------------------------------------------------------------
__global__ __launch_bounds__(128) void k_gemm_wmma_f32(
        const float* __restrict__ A, const float* __restrict__ W,
        float* __restrict__ C, int M, int K, int N) {
    __shared__ float sW[KC * LDSW];   // 20 KB of the 320 KB WGP LDS

    const int tid  = threadIdx.x;
    const int wave = tid >> 5;
    const int lane = tid & 31;
    const int half = lane >> 4;       // 0: lanes 0-15, 1: lanes 16-31
    const int l16  = lane & 15;
    const int m0 = (blockIdx.x * 4 + wave) * 16;
    const int n0 = blockIdx.y * 64;
    const bool active = (m0 < M);     // wave-uniform; idle waves still hit barriers

    v8f acc0 = {}, acc1 = {}, acc2 = {}, acc3 = {};
    const float* arow = A + (size_t)((active ? m0 : 0) + l16) * K + half * 2;

    for (int kc = 0; kc < K; kc += KC) {
        // --- cooperative stage: W[kc..kc+KC)[n0..n0+64) -> LDS (float4 lanes) ---
        for (int i = tid; i < KC * 16; i += 128) {
            int r  = i >> 4;          // row within chunk
            int c4 = i & 15;          // float4 column
            float4 v = *(const float4*)(W + (size_t)(kc + r) * N + n0 + c4 * 4);
            *(float4*)(sW + r * LDSW + c4 * 4) = v;
        }
        __syncthreads();

        if (active) {
            for (int k = 0; k < KC; k += 4) {
                __builtin_prefetch(arow + kc + k + 64, 0, 3);  // global_prefetch_b8
                v2f a;
                a.x = arow[kc + k];
                a.y = arow[kc + k + 1];
                const float* s0 = sW + (k + half * 2) * LDSW + l16;  // B VGPR0 row
                const float* s1 = s0 + LDSW;                          // B VGPR1 row
                v2f b0, b1, b2, b3;
                b0.x = s0[0];  b0.y = s1[0];
                b1.x = s0[16]; b1.y = s1[16];
                b2.x = s0[32]; b2.y = s1[32];
                b3.x = s0[48]; b3.y = s1[48];
                acc0 = __builtin_amdgcn_wmma_f32_16x16x4_f32(false, a, false, b0, (short)0, acc0, false, false);
                acc1 = __builtin_amdgcn_wmma_f32_16x16x4_f32(false, a, false, b1, (short)0, acc1, false, false);
                acc2 = __builtin_amdgcn_wmma_f32_16x16x4_f32(false, a, false, b2, (short)0, acc2, false, false);
                acc3 = __builtin_amdgcn_wmma_f32_16x16x4_f32(false, a, false, b3, (short)0, acc3, false, false);
            }
        }
        __syncthreads();
    }

    if (active) {
        // C/D layout: VGPR v -> row m0 + half*8 + v, col n0 + tile + l16
        float* crow = C + (size_t)(m0 + half * 8) * N + n0 + l16;
#pragma unroll
        for (int v = 0; v < 8; ++v) {
            crow[0]  = acc0[v];
            crow[16] = acc1[v];
            crow[32] = acc2[v];
            crow[48] = acc3[v];
            crow += N;
        }
    }
}

// ---------------------------------------------------------------------------
// GCN aggregation:  OUT[n] = sum_{e: dst=n} XW[src]*dis[src]*dis[dst]  (+ self loop)
// Both feature buffers (51 MB) are resident in the 192 MB L2 -> atomic-rate bound.
// ---------------------------------------------------------------------------
__global__ void k_agg_self(const float* __restrict__ XW, const float* __restrict__ dis,
                           float* __restrict__ OUT, int n_nodes) {
    int idx = blockIdx.x * blockDim.x + threadIdx.x;        // node * (HID/4)
    if (idx >= n_nodes * (HID / 4)) return;
    int n  = idx >> 6;
    int f4 = idx & 63;
    float d = dis[n];
    float nn = d * d;                                        // self-loop norm
    float4 v = ((const float4*)(XW + (size_t)n * HID))[f4];
    float4 o = make_float4(v.x * nn, v.y * nn, v.z * nn, v.w * nn);
    ((float4*)(OUT + (size_t)n * HID))[f4] = o;
}

__global__ void k_agg_edges(const int* __restrict__ src, const int* __restrict__ dst,
                            const float* __restrict__ dis, const float* __restrict__ XW,
                            float* __restrict__ OUT, int n_edges) {
    int idx = blockIdx.x * blockDim.x + threadIdx.x;        // edge * (HID/4)
    if (idx >= n_edges * (HID / 4)) return;                  // 51.2M < 2^31
    int e  = idx >> 6;
    int f4 = idx & 63;
    int s = src[e], d = dst[e];
    float nrm = dis[s] * dis[d];
    float4 v = ((const float4*)(XW + (size_t)s * HID))[f4];  // L2-resident gather
    float* o = OUT + (size_t)d * HID + f4 * 4;
    atomicAdd(o + 0, v.x * nrm);
    atomicAdd(o + 1, v.y * nrm);
    atomicAdd(o + 2, v.z * nrm);
    atomicAdd(o + 3, v.w * nrm);
}

__global__ void k_bias_relu(const float* __restrict__ IN, const float* __restrict__ b,
                            float* __restrict__ OUT, int n_nodes) {
    int idx = blockIdx.x * blockDim.x + threadIdx.x;
    if (idx >= n_nodes * (HID / 4)) return;
    int n  = idx >> 6;
    int f4 = idx & 63;
    float4 v  = ((const float4*)(IN + (size_t)n * HID))[f4];
    float4 bb = ((const float4*)b)[f4];
    float4 o  = make_float4(fmaxf(v.x + bb.x, 0.0f), fmaxf(v.y + bb.y, 0.0f),
                            fmaxf(v.z + bb.z, 0.0f), fmaxf(v.w + bb.w, 0.0f));
    ((float4*)(OUT + (size_t)n * HID))[f4] = o;
}

__global__ void k_pool(const float* __restrict__ H, const int* __restrict__ batch,
                       float* __restrict__ pooled, int n_nodes) {
    int idx = blockIdx.x * blockDim.x + threadIdx.x;
    if (idx >= n_nodes * (HID / 4)) return;
    int n  = idx >> 6;
    int f4 = idx & 63;
    int g = batch[n];
    float4 v = ((const float4*)(H + (size_t)n * HID))[f4];
    float* o = pooled + (size_t)g * HID + f4 * 4;
    atomicAdd(o + 0, v.x);
    atomicAdd(o + 1, v.y);
    atomicAdd(o + 2, v.z);
    atomicAdd(o + 3, v.w);
}

__global__ void k_fc(const float* __restrict__ pooled, const float* __restrict__ counts,
                     const float* __restrict__ action, const float* __restrict__ fcW,
                     const float* __restrict__ fcb, float* __restrict__ out) {
    int g = blockIdx.x * blockDim.x + threadIdx.x;
    if (g >= NGRAPHS) return;
    float inv = 1.0f / counts[g];
    float acc = fcb[0];
    const float* pg = pooled + (size_t)g * HID;
    for (int f = 0; f < HID; ++f) acc += pg[f] * inv * fcW[f];
    const float* ag = action + (size_t)g * ACTD;
    for (int a = 0; a < ACTD; ++a) acc += ag[a] * fcW[HID + a];
    out[g] = acc;
}

// ---------------------------------------------------------------------------
// Launcher
// ---------------------------------------------------------------------------
extern "C" void kernel_launch(void* const* d_in, const int* in_sizes, int n_in,
                              void* d_out, int out_size, void* d_ws, size_t ws_size,
                              hipStream_t stream) {
    const float* x      = (const float*)d_in[0];
    const int*   ei     = (const int*)d_in[1];
    const int*   src    = ei;
    const int*   dst    = ei + NEDGES;
    const int*   batch  = (const int*)d_in[2];
    const float* action = (const float*)d_in[3];
    float* out = (float*)d_out;

    // Workspace carve-up (floats)
    float* ws = (float*)d_ws;
    float* dis    = ws;                       // 50000 -> pad 50176
    float* counts = ws + 50176;               // 64
    float* pooled = ws + 50432;               // 16384 -> pad to 66816
    float* buf0   = ws + 66816;               // 12,800,000
    float* buf1   = ws + 66816 + (size_t)NNODES * HID;

    const int T = 256;
    const int bNodes = (NNODES + T - 1) / T;
    const int bEdges = (NEDGES + T - 1) / T;
    const int bNF    = (NNODES * (HID / 4) + T - 1) / T;   // 12,500
    const int bEF    = (NEDGES * (HID / 4) + T - 1) / T;   // 200,000

    // Degree (self loops contribute 1) -> dis = rsqrt(deg); graph counts
    k_fill<<<bNodes, T, 0, stream>>>(dis, 1.0f, NNODES);
    k_deg_edges<<<bEdges, T, 0, stream>>>(dst, dis, NEDGES);
    k_rsqrt_inplace<<<bNodes, T, 0, stream>>>(dis, NNODES);
    k_fill<<<1, 64, 0, stream>>>(counts, 0.0f, NGRAPHS);
    k_counts<<<bNodes, T, 0, stream>>>(batch, counts, NNODES);

    // GEMM grids: 3125 row-tiles / 4 waves per block; 256 cols / 64 per wave-strip
    dim3 gemmGrid((NNODES / 16 + 3) / 4, HID / 64);

    for (int head = 0; head < 2; ++head) {
        const float* W1  = (const float*)d_in[4 + head * 6];
        const float* b1  = (const float*)d_in[5 + head * 6];
        const float* W2  = (const float*)d_in[6 + head * 6];
        const float* b2  = (const float*)d_in[7 + head * 6];
        const float* fcW = (const float*)d_in[8 + head * 6];
        const float* fcb = (const float*)d_in[9 + head * 6];

        // Layer 1: buf0 = x @ W1; buf1 = aggregate(buf0); buf0 = relu(buf1 + b1)
        k_gemm_wmma_f32<<<gemmGrid, 128, 0, stream>>>(x, W1, buf0, NNODES, FEAT, HID);
        k_agg_self<<<bNF, T, 0, stream>>>(buf0, dis, buf1, NNODES);
        k_agg_edges<<<bEF, T, 0, stream>>>(src, dst, dis, buf0, buf1, NEDGES);
        k_bias_relu<<<bNF, T, 0, stream>>>(buf1, b1, buf0, NNODES);

        // Layer 2: buf1 = buf0 @ W2; buf0 = aggregate(buf1); buf1 = relu(buf0 + b2)
        k_gemm_wmma_f32<<<gemmGrid, 128, 0, stream>>>(buf0, W2, buf1, NNODES, HID, HID);
        k_agg_self<<<bNF, T, 0, stream>>>(buf1, dis, buf0, NNODES);
        k_agg_edges<<<bEF, T, 0, stream>>>(src, dst, dis, buf1, buf0, NEDGES);
        k_bias_relu<<<bNF, T, 0, stream>>>(buf0, b2, buf1, NNODES);

        // Mean pool + FC head
        k_fill<<<(NGRAPHS * HID + T - 1) / T, T, 0, stream>>>(pooled, 0.0f, NGRAPHS * HID);
        k_pool<<<bNF, T, 0, stream>>>(buf1, batch, pooled, NNODES);
        k_fc<<<1, 64, 0, stream>>>(pooled, counts, action, fcW, fcb, out + head * NGRAPHS);
    }
}